// DifferentiablePaletteQuantization_18116172055276
// MI455X (gfx1250) — compile-verified
//
#include <hip/hip_runtime.h>

typedef __attribute__((ext_vector_type(2))) float v2f;
typedef __attribute__((ext_vector_type(8))) float v8f;

#define LOG2E 1.4426950408889634f

// One wave processes a tile of 16 pixels per iteration.
//   Stage 1 (transposed): S^T[64x16] = P[64x4(rgb0)] * X^T[4x16]
//                         via 4x V_WMMA_F32_16X16X4_F32 (A = palette chunk, B = pixels)
//   C-layout then gives: lane l -> pixel n = l%16, VGPR v -> entry 16t + v + 8*(l/16).
//   Each lane softmax-accumulates its own 32 entries (4 scalar accumulators),
//   one shfl_xor(16) merges the complementary 32 entries from lane l^16.
__global__ __launch_bounds__(256) void palette_quant_wmma_kernel(
    const float* __restrict__ images,    // [b, npix, 3]
    const float* __restrict__ palettes,  // [b, 64, 3]
    const float* __restrict__ temperature,
    float* __restrict__ out,             // [b, npix, 3]
    int npixPerBatch, int tilesPerBatch, int wavesPerBatch, int blocksPerBatch)
{
    __shared__ float4 palTab[64];   // {r, g, b, -||p||^2 * invT * LOG2E}

    const int lane        = threadIdx.x & 31;
    const int waveInBlock = threadIdx.x >> 5;
    const int batch       = blockIdx.x / blocksPerBatch;
    const int blkInBatch  = blockIdx.x % blocksPerBatch;
    const int wave        = blkInBatch * 8 + waveInBlock;

    const int  ln = lane & 15;
    const bool lo = lane < 16;

    const float invT = 1.0f / temperature[0];
    const float kA   = 2.0f * invT * LOG2E;   // scale on s = p.x inside exp2

    const float* pal = palettes + (size_t)batch * 64 * 3;

    // Build the per-entry constant table once per block (1 KB LDS).
    if (threadIdx.x < 64) {
        const float r  = pal[threadIdx.x * 3 + 0];
        const float g  = pal[threadIdx.x * 3 + 1];
        const float b  = pal[threadIdx.x * 3 + 2];
        const float pp = r * r + g * g + b * b;
        palTab[threadIdx.x] = make_float4(r, g, b, -pp * invT * LOG2E);
    }
    __syncthreads();

    // A chunks: palette entries in A-layout (16x4 f32), hoisted per wave.
    // lanes 0-15: (K0,K1) = (r,g) of entry 16t+ln ; lanes 16-31: (K2,K3) = (b,0)
    v2f At[4];
#pragma unroll
    for (int t = 0; t < 4; ++t) {
        const int e = t * 16 + ln;
        const float r = pal[e * 3 + 0], g = pal[e * 3 + 1], b = pal[e * 3 + 2];
        At[t].x = lo ? r : b;
        At[t].y = lo ? g : 0.0f;
    }

    // Per-lane LDS base: entry(t,v,h) = 16t + v + 8h  ->  byte offset h*128 + (16t+v)*16
    const float4* tabH = (const float4*)((const char*)palTab + ((lane & 16) << 3));

    const float* imgB = images + (size_t)batch * npixPerBatch * 3;
    float*       outB = out    + (size_t)batch * npixPerBatch * 3;

    for (int tile = wave; tile < tilesPerBatch; tile += wavesPerBatch) {
        const int    pix0 = tile * 16;
        const float* ip   = imgB + (size_t)pix0 * 3;

        // prefetch this wave's next tile (global_prefetch_b8)
        __builtin_prefetch(ip + (size_t)wavesPerBatch * 48, 0, 1);

        // B = X^T (4 x 16 pixels): lane n: rows (0,1)=(r,g) / rows (2,3)=(b,0)
        const float x0 = ip[ln * 3 + (lo ? 0 : 2)];
        const float x1 = ip[ln * 3 + 1];
        v2f B;
        B.x = x0;
        B.y = lo ? x1 : 0.0f;

        v8f cz = {};
        const v8f C0 = __builtin_amdgcn_wmma_f32_16x16x4_f32(false, At[0], false, B, (short)0, cz, false, false);
        const v8f C1 = __builtin_amdgcn_wmma_f32_16x16x4_f32(false, At[1], false, B, (short)0, cz, false, false);
        const v8f C2 = __builtin_amdgcn_wmma_f32_16x16x4_f32(false, At[2], false, B, (short)0, cz, false, false);
        const v8f C3 = __builtin_amdgcn_wmma_f32_16x16x4_f32(false, At[3], false, B, (short)0, cz, false, false);

        // Each lane: softmax-weighted accumulation over its 32 entries.
        // w = exp((2*s - ||p||^2)/T); the per-pixel ||x||^2 term cancels in softmax.
        float Z = 0.f, R = 0.f, G = 0.f, Bl = 0.f;
#pragma unroll
        for (int t = 0; t < 4; ++t) {
            const v8f C = (t == 0) ? C0 : (t == 1) ? C1 : (t == 2) ? C2 : C3;
#pragma unroll
            for (int v = 0; v < 8; ++v) {
                const float4 p = tabH[t * 16 + v];
                const float  w = __builtin_amdgcn_exp2f(fmaf(C[v], kA, p.w));
                Z += w;
                R  = fmaf(w, p.x, R);
                G  = fmaf(w, p.y, G);
                Bl = fmaf(w, p.z, Bl);
            }
        }

        // Merge the complementary 32 entries held by lane l^16.
        Z  += __shfl_xor(Z, 16, 32);
        R  += __shfl_xor(R, 16, 32);
        G  += __shfl_xor(G, 16, 32);
        Bl += __shfl_xor(Bl, 16, 32);

        const float invZ = __builtin_amdgcn_rcpf(Z);
        if (lo) {
            float* op = outB + (size_t)(pix0 + ln) * 3;
            op[0] = R  * invZ;
            op[1] = G  * invZ;
            op[2] = Bl * invZ;
        }
    }
}

extern "C" void kernel_launch(void* const* d_in, const int* in_sizes, int n_in,
                              void* d_out, int out_size, void* d_ws, size_t ws_size,
                              hipStream_t stream) {
    const float* images      = (const float*)d_in[0];
    const float* palettes    = (const float*)d_in[1];
    const float* temperature = (const float*)d_in[2];
    float*       out         = (float*)d_out;

    const int palElems = in_sizes[1];            // b * 64 * 3
    const int b        = palElems / (64 * 3);    // 16
    const int npix     = in_sizes[0] / (b * 3);  // 65536
    const int tiles    = npix / 16;              // 4096 tiles of 16 pixels

    // ~8 tiles per wave, 8 waves (256 threads) per block
    int blocksPerBatch = (tiles + 63) / 64;
    if (blocksPerBatch < 1) blocksPerBatch = 1;
    const int wavesPerBatch = blocksPerBatch * 8;

    dim3 grid(b * blocksPerBatch), block(256);
    hipLaunchKernelGGL(palette_quant_wmma_kernel, grid, block, 0, stream,
                       images, palettes, temperature, out,
                       npix, tiles, wavesPerBatch, blocksPerBatch);
}